// channelAttention_fusion_85684597555339
// MI455X (gfx1250) — compile-verified
//
#include <hip/hip_runtime.h>
#include <hip/hip_bf16.h>

// Problem constants (from reference): B=32, C=768, H=W=56
#define BATCH   32
#define CH      768
#define CH2     1536          // 2*C
#define HW      3136          // 56*56
#define HW4     784           // HW/4
#define BN_EPS  1e-5f

typedef __attribute__((ext_vector_type(2))) float v2f;
typedef __attribute__((ext_vector_type(8))) float v8f;

// ---------------------------------------------------------------------------
// Kernel 1: global avg+max pool over HW for each (b,c) row.
// im[b, 0:768] = avg, im[b, 768:1536] = max   (matches jnp.concatenate order)
// ---------------------------------------------------------------------------
__global__ void ca_pool_kernel(const float* __restrict__ in, float* __restrict__ im) {
    __shared__ float s_sum[256];
    __shared__ float s_max[256];
    const int row = blockIdx.x;                    // b*C + c
    const float4* p = (const float4*)(in + (size_t)row * HW);

    float sum = 0.0f;
    float mx  = -__builtin_inff();
    for (int i = threadIdx.x; i < HW4; i += 256) { // 784 float4 per row
        float4 v = p[i];
        sum += (v.x + v.y) + (v.z + v.w);
        mx = fmaxf(mx, fmaxf(fmaxf(v.x, v.y), fmaxf(v.z, v.w)));
    }
    s_sum[threadIdx.x] = sum;
    s_max[threadIdx.x] = mx;
    __syncthreads();
    #pragma unroll
    for (int s = 128; s > 0; s >>= 1) {
        if (threadIdx.x < s) {
            s_sum[threadIdx.x] += s_sum[threadIdx.x + s];
            s_max[threadIdx.x] = fmaxf(s_max[threadIdx.x], s_max[threadIdx.x + s]);
        }
        __syncthreads();
    }
    if (threadIdx.x == 0) {
        const int b = row / CH;
        const int c = row % CH;
        im[(size_t)b * CH2 + c]       = s_sum[0] * (1.0f / (float)HW);
        im[(size_t)b * CH2 + CH + c]  = s_max[0];
    }
}

// ---------------------------------------------------------------------------
// Kernel 2: im2 = BN( im @ W^T + bias )   via V_WMMA_F32_16X16X4_F32.
//   M = 32 (batch), N = 768 (out channels), K = 1536.
//   One wave per 16x16 output tile; 384 chained WMMAs over K.
// A frag (16x4 f32, 2 VGPRs): lane L -> row m = L&15; VGPR v holds K = k + 2*(L>>4) + v
// B frag (4x16 f32, 2 VGPRs): lane L -> col n = L&15; VGPR v holds K = k + 2*(L>>4) + v
//   B[k][n] = W^T[k][n] = conv_w[n*1536 + k]  (conv_w is [768, 1536])
// C/D (16x16 f32, 8 VGPRs): acc[r] = D[m = r + 8*(L>>4)][n = L&15]
// ---------------------------------------------------------------------------
__global__ void ca_gemm_bn_kernel(const float* __restrict__ im,     // [32,1536]
                                  const float* __restrict__ w,      // [768,1536]
                                  const float* __restrict__ bias,   // [768]
                                  const float* __restrict__ gamma,
                                  const float* __restrict__ beta,
                                  const float* __restrict__ mean,
                                  const float* __restrict__ var,
                                  float* __restrict__ im2) {        // [32,768]
    const int mt   = blockIdx.x & 1;        // 0..1   (M tiles: 32/16)
    const int nt   = blockIdx.x >> 1;       // 0..47  (N tiles: 768/16)
    const int lane = threadIdx.x;           // 0..31 (wave32)
    const int half = lane >> 4;             // K-half select
    const int l16  = lane & 15;

    const float* arow = im + (size_t)(mt * 16 + l16) * CH2;  // A row m
    const float* brow = w  + (size_t)(nt * 16 + l16) * CH2;  // W row n (== W^T col n)

    v8f acc = {};
    for (int k = 0; k < CH2; k += 4) {
        v2f a = *(const v2f*)(arow + k + 2 * half);   // K = k+2h, k+2h+1
        v2f b = *(const v2f*)(brow + k + 2 * half);
        // 8 args: (neg_a, A, neg_b, B, c_mod, C, reuse_a, reuse_b)
        acc = __builtin_amdgcn_wmma_f32_16x16x4_f32(
            false, a, false, b, (short)0, acc, false, false);
    }

    const int n = nt * 16 + l16;                      // out channel (per-lane const)
    const float scale = gamma[n] * rsqrtf(var[n] + BN_EPS);
    const float shift = beta[n] - mean[n] * scale;
    const float bsum  = bias[n];
    #pragma unroll
    for (int r = 0; r < 8; ++r) {
        const int m = mt * 16 + r + 8 * half;         // batch index
        im2[(size_t)m * CH + n] = (acc[r] + bsum) * scale + shift;
    }
}

// ---------------------------------------------------------------------------
// Kernel 3: att[b,p] = sum_c im2[b,c] * in[b,c,p].
// One block per (b, 256-pixel chunk); im2 row cached in LDS; per-c loads are
// fully coalesced across the 256 consecutive pixels.
// ---------------------------------------------------------------------------
#define ATT_CHUNKS 13   // ceil(3136/256)

__global__ void ca_att_kernel(const float* __restrict__ in,
                              const float* __restrict__ im2,
                              float* __restrict__ att) {
    __shared__ float s_im[CH];
    const int b     = blockIdx.x / ATT_CHUNKS;
    const int chunk = blockIdx.x % ATT_CHUNKS;

    for (int i = threadIdx.x; i < CH; i += 256)
        s_im[i] = im2[(size_t)b * CH + i];
    __syncthreads();

    const int pix = chunk * 256 + threadIdx.x;
    if (pix >= HW) return;

    const float* p = in + (size_t)b * CH * HW + pix;
    float acc = 0.0f;
    #pragma unroll 4
    for (int c = 0; c < CH; ++c) {
        if ((c & 63) == 0 && c + 64 < CH)
            __builtin_prefetch(p + (size_t)(c + 64) * HW, 0, 0);  // global_prefetch_b8
        acc += s_im[c] * p[(size_t)c * HW];
    }
    att[(size_t)b * HW + pix] = acc;
}

// ---------------------------------------------------------------------------
// Kernel 4: out[b,c,p] = att[b,p] * in[b,c,p], float4 streaming (b128).
// Total float4 = 32*768*784 = 19,267,584 = 75264 blocks * 256 threads exactly.
// ---------------------------------------------------------------------------
__global__ void ca_gate_kernel(const float* __restrict__ in,
                               const float* __restrict__ att,
                               float* __restrict__ out) {
    const size_t i4 = (size_t)blockIdx.x * 256 + threadIdx.x;
    const int bc = (int)(i4 / HW4);       // b*C + c
    const int p4 = (int)(i4 % HW4);
    const int b  = bc / CH;

    float4 v = ((const float4*)in)[i4];
    float4 a = ((const float4*)att)[(size_t)b * HW4 + p4];
    float4 o;
    o.x = a.x * v.x; o.y = a.y * v.y; o.z = a.z * v.z; o.w = a.w * v.w;
    ((float4*)out)[i4] = o;
}

// ---------------------------------------------------------------------------
extern "C" void kernel_launch(void* const* d_in, const int* in_sizes, int n_in,
                              void* d_out, int out_size, void* d_ws, size_t ws_size,
                              hipStream_t stream) {
    const float* in    = (const float*)d_in[0];  // [32,768,56,56]
    const float* w     = (const float*)d_in[1];  // [768,1536]
    const float* bias  = (const float*)d_in[2];  // [768]
    const float* gamma = (const float*)d_in[3];
    const float* beta  = (const float*)d_in[4];
    const float* mean  = (const float*)d_in[5];
    const float* var   = (const float*)d_in[6];
    float* out = (float*)d_out;

    // Workspace layout (all fully written before read; ~0.7 MB total)
    float* im  = (float*)d_ws;                       // [32,1536]
    float* im2 = im  + (size_t)BATCH * CH2;          // [32,768]
    float* att = im2 + (size_t)BATCH * CH;           // [32,3136]

    ca_pool_kernel<<<BATCH * CH, 256, 0, stream>>>(in, im);
    ca_gemm_bn_kernel<<<2 * (CH / 16), 32, 0, stream>>>(im, w, bias, gamma, beta,
                                                        mean, var, im2);
    ca_att_kernel<<<BATCH * ATT_CHUNKS, 256, 0, stream>>>(in, im2, att);
    ca_gate_kernel<<<(BATCH * CH * HW4) / 256, 256, 0, stream>>>(in, att, out);
}